// InputProjection_W_TimeSensor_Masking_33535104647573
// MI455X (gfx1250) — compile-verified
//
#include <hip/hip_runtime.h>

// InputProjection with time/sensor masking + fused LayerNorm, MI455X (gfx1250).
// Strategy: per-batch premasked B-matrix in LDS + constant rows for masked cases,
// f32 WMMA (16x16x4) for the [T x 51] @ [51 x 64] GEMM, LayerNorm fused via
// half-wave shfl_xor butterflies over the WMMA accumulator layout.

#define B_   64
#define T_   4096
#define C_   51
#define D_   64
#define KP_  52            // C padded to multiple of 4
#define KSTEPS 13          // KP_/4
#define ROWS_PER_BLOCK 128 // 8 waves * 16 rows
#define LN_EPS 1e-5f

typedef __attribute__((ext_vector_type(2))) float v2f;
typedef __attribute__((ext_vector_type(8))) float v8f;

__global__ __launch_bounds__(256)
void InputProjection_W_TimeSensor_Masking_33535104647573_kernel(
    const float* __restrict__ x,        // [B,T,C]
    const float* __restrict__ W,        // [C,D]
    const float* __restrict__ Wm,       // [C,D]
    const float* __restrict__ gamma,    // [D]
    const float* __restrict__ beta,     // [D]
    const int*   __restrict__ time_mask,   // [B,T]
    const int*   __restrict__ sensor_mask, // [B,C]
    float* __restrict__ out)            // [B,T,D]
{
    // B-matrix staged as K-pairs so each lane's 2 B values are one ds_load_b64:
    // Bsh[kp][d][s] = W_masked[2*kp + s][d]   (rows >= C zero-padded)
    __shared__ float Bsh[KP_ / 2][D_][2];
    __shared__ float smrow[D_];   // sum of Wm rows where sensor_mask==1
    __shared__ float allrow[D_];  // sum of all Wm rows (fully time-masked rows)
    __shared__ int   tmsh[ROWS_PER_BLOCK];
    __shared__ int   smsh[KP_];

    const int b       = blockIdx.y;
    const int rowBase = blockIdx.x * ROWS_PER_BLOCK;
    const int tid     = threadIdx.x;

    // ---- per-block setup -------------------------------------------------
    if (tid < KP_)            smsh[tid] = (tid < C_) ? sensor_mask[b * C_ + tid] : 1;
    if (tid < ROWS_PER_BLOCK) tmsh[tid] = time_mask[(size_t)b * T_ + rowBase + tid];
    __syncthreads();

    // Premasked, zero-padded B matrix: 52*64 elems, 13 per thread.
    for (int i = tid; i < KP_ * D_; i += 256) {
        const int c = i >> 6;
        const int d = i & 63;
        float w = 0.0f;
        if (c < C_ && smsh[c] == 0) w = W[c * D_ + d];
        Bsh[c >> 1][d][c & 1] = w;
    }
    // Constant rows (one thread per output column).
    if (tid < D_) {
        float s = 0.0f, a = 0.0f;
        for (int c = 0; c < C_; ++c) {
            const float wm = Wm[c * D_ + tid];
            a += wm;
            if (smsh[c] != 0) s += wm;
        }
        smrow[tid]  = s;
        allrow[tid] = a;
    }
    __syncthreads();

    // ---- WMMA main loop: wave -> 16 rows x 64 cols -----------------------
    const int wave = tid >> 5;
    const int lane = tid & 31;
    const int half = lane >> 4;   // lane group: 0 -> lanes 0-15, 1 -> 16-31
    const int m    = lane & 15;
    const int r0   = rowBase + wave * 16;

    // A-matrix 16x4 f32 layout: lane m = row M, VGPR v holds K = 2*half + v.
    const float* __restrict__ xrow = x + ((size_t)b * T_ + (r0 + m)) * C_;

    v8f acc[4];
    const v8f vzero = {};
    #pragma unroll
    for (int n = 0; n < 4; ++n) acc[n] = vzero;

    #pragma unroll
    for (int kk = 0; kk < KSTEPS; ++kk) {
        const int k0 = 4 * kk + 2 * half;
        v2f a;
        a.x = xrow[k0];
        a.y = (k0 + 1 < C_) ? xrow[k0 + 1] : 0.0f;   // zero-pad K=51
        const int kp = 2 * kk + half;
        #pragma unroll
        for (int n = 0; n < 4; ++n) {
            const v2f bf = *(const v2f*)&Bsh[kp][n * 16 + m][0];
            acc[n] = __builtin_amdgcn_wmma_f32_16x16x4_f32(
                         false, a, false, bf, (short)0, acc[n], false, false);
        }
    }

    // ---- fused epilogue: mask select + LayerNorm + store -----------------
    float gam[4], bet[4];
    #pragma unroll
    for (int n = 0; n < 4; ++n) {
        gam[n] = gamma[n * 16 + m];
        bet[n] = beta [n * 16 + m];
    }

    // C/D layout: VGPR j, lanes 0-15 -> row j, lanes 16-31 -> row j+8; col = n*16+m.
    // A full 64-wide output row lives in one 16-lane half across the 4 acc regs.
    #pragma unroll
    for (int j = 0; j < 8; ++j) {
        const int lr  = wave * 16 + j + 8 * half;
        const int row = rowBase + lr;
        const int tmf = tmsh[lr];

        float v[4];
        float s = 0.0f, q = 0.0f;
        #pragma unroll
        for (int n = 0; n < 4; ++n) {
            float p = acc[n][j];
            p = tmf ? allrow[n * 16 + m] : (p + smrow[n * 16 + m]);
            v[n] = p;
            s += p;
            q += p * p;
        }
        // Reduce across the 16 lanes of this half (xor 1,2,4,8 stays in-half).
        #pragma unroll
        for (int off = 1; off < 16; off <<= 1) {
            s += __shfl_xor(s, off, 32);
            q += __shfl_xor(q, off, 32);
        }
        const float mean = s * (1.0f / 64.0f);
        const float var  = q * (1.0f / 64.0f) - mean * mean;
        const float inv  = rsqrtf(var + LN_EPS);

        float* __restrict__ orow = out + ((size_t)b * T_ + row) * D_;
        #pragma unroll
        for (int n = 0; n < 4; ++n)
            orow[n * 16 + m] = (v[n] - mean) * inv * gam[n] + bet[n];
    }
}

extern "C" void kernel_launch(void* const* d_in, const int* in_sizes, int n_in,
                              void* d_out, int out_size, void* d_ws, size_t ws_size,
                              hipStream_t stream) {
    (void)in_sizes; (void)n_in; (void)out_size; (void)d_ws; (void)ws_size;
    const float* x     = (const float*)d_in[0];
    const float* W     = (const float*)d_in[1];
    const float* Wm    = (const float*)d_in[2];
    const float* gamma = (const float*)d_in[3];
    const float* beta  = (const float*)d_in[4];
    const int*   tm    = (const int*)d_in[5];
    const int*   sm    = (const int*)d_in[6];

    dim3 grid(T_ / ROWS_PER_BLOCK, B_);  // (32, 64)
    InputProjection_W_TimeSensor_Masking_33535104647573_kernel
        <<<grid, 256, 0, stream>>>(x, W, Wm, gamma, beta, tm, sm, (float*)d_out);
}